// DecoderCell_75187697484052
// MI455X (gfx1250) — compile-verified
//
#include <hip/hip_runtime.h>
#include <hip/hip_bf16.h>
#include <math.h>

#define B_     64
#define S_     128
#define VOCAB_ 32000
#define EMB_   256
#define UNITS_ 512
#define H_     1024
#define DENC_  1024
#define IN_    1280   // EMB + D_ENC

typedef __attribute__((ext_vector_type(2))) float v2f;
typedef __attribute__((ext_vector_type(8))) float v8f;
typedef __attribute__((ext_vector_type(4))) unsigned int v4u;
typedef __attribute__((ext_vector_type(8))) int v8i_;
typedef __attribute__((ext_vector_type(4))) int v4i_;

__device__ __forceinline__ float sigm_(float x) { return 1.0f / (1.0f + expf(-x)); }

// Issue one TDM load of a 2-D tile (elem = f32) into LDS offset 0.
//   tile = tile_d0 x tile_d1 elements, row stride = stride0 elements.
// Caller must be the wave that later waits on TENSORcnt.
__device__ __forceinline__ void tdm_load_f32_tile(const float* gsrc,
                                                  unsigned int tensor_d0, unsigned int tensor_d1,
                                                  unsigned int tile_d0,  unsigned int tile_d1,
                                                  unsigned long long stride0) {
    unsigned long long ga = (unsigned long long)(const void*)gsrc;
    v4u g0;
    g0[0] = 1u;                                            // count=1, user mode
    g0[1] = 0u;                                            // lds_addr = 0 (tile at LDS base)
    g0[2] = (unsigned int)(ga & 0xFFFFFFFFu);              // global_addr[31:0]
    g0[3] = (unsigned int)((ga >> 32) & 0x01FFFFFFu) | (2u << 30);  // addr[56:32] | type=2
    v8i_ g1;
    g1[0] = (int)(2u << 16);                               // data_size=4B; no mask/pad/iter
    g1[1] = (int)((tensor_d0 & 0xFFFFu) << 16);            // [47:32]=bar addr(0) | dim0 lo16
    g1[2] = (int)(((tensor_d0 >> 16) & 0xFFFFu) | ((tensor_d1 & 0xFFFFu) << 16));
    g1[3] = (int)(((tensor_d1 >> 16) & 0xFFFFu) | ((tile_d0 & 0xFFFFu) << 16));
    g1[4] = (int)(tile_d1 & 0xFFFFu);                      // tile_dim1 | tile_dim2=0
    g1[5] = (int)(unsigned int)(stride0 & 0xFFFFFFFFull);  // dim0_stride lo32
    g1[6] = (int)(unsigned int)((stride0 >> 32) & 0xFFFFull); // dim0_stride hi16 | dim1_stride lo16=0
    g1[7] = 0;                                             // dim1_stride hi32 (unused, 2-D tile)
    v4i_ g2 = {};
    v4i_ g3 = {};
#if defined(__clang_major__) && (__clang_major__ >= 23)
    v8i_ g4 = {};
    __builtin_amdgcn_tensor_load_to_lds(g0, g1, g2, g3, g4, 0);
#else
    __builtin_amdgcn_tensor_load_to_lds(g0, g1, g2, g3, 0);
#endif
}

// ---------------------------------------------------------------- q = h @ W2
__global__ void q_proj_kernel(const float* __restrict__ h, const float* __restrict__ W2,
                              float* __restrict__ q) {
    const int b = blockIdx.x;          // 64
    const int u = threadIdx.x;         // 512
    const float* hr = h + b * H_;
    float acc = 0.0f;
    for (int k = 0; k < H_; ++k) acc += hr[k] * W2[k * UNITS_ + u];
    q[b * UNITS_ + u] = acc;
}

// ------------- fused score: tanh(enc@W1 + q) @ V   (one wave per 16-row tile)
__global__ void attn_score_kernel(const float* __restrict__ enc, const float* __restrict__ W1,
                                  const float* __restrict__ V,  const float* __restrict__ q,
                                  float* __restrict__ scores) {
    __shared__ float lA[16 * DENC_];   // 64 KB: 16 rows of enc_out, reused for 32 N-tiles
    const int tile = blockIdx.x;       // 512 tiles of 16 (b,s) rows (S%16==0 -> b uniform)
    const int lane = threadIdx.x;      // wave32
    const int row0 = tile * 16;
    const int b    = row0 / S_;

    // TDM: one 1-D DMA of the contiguous 16x1024 f32 A-tile into LDS.
    if (lane == 0) {
        tdm_load_f32_tile(enc + (size_t)row0 * DENC_,
                          /*tensor_d0=*/16 * DENC_, /*tensor_d1=*/1u,
                          /*tile_d0=*/16 * DENC_,   /*tile_d1=*/1u,
                          /*stride0=*/16 * DENC_);
        __builtin_amdgcn_s_wait_tensorcnt(0);
    }
    __syncthreads();

    const int lo = lane & 15;
    const int hi = lane >> 4;
    float part[8];
#pragma unroll
    for (int r = 0; r < 8; ++r) part[r] = 0.0f;

    for (int nt = 0; nt < UNITS_ / 16; ++nt) {
        const int col = nt * 16 + lo;
        v8f acc = {};
        for (int k0 = 0; k0 < DENC_; k0 += 4) {
            const int ka = k0 + 2 * hi;
            v2f a, bb;
            a.x  = lA[lo * DENC_ + ka];
            a.y  = lA[lo * DENC_ + ka + 1];
            bb.x = W1[(size_t)ka * UNITS_ + col];
            bb.y = W1[(size_t)(ka + 1) * UNITS_ + col];
            acc = __builtin_amdgcn_wmma_f32_16x16x4_f32(false, a, false, bb,
                                                        (short)0, acc, false, false);
        }
        const float qv = q[b * UNITS_ + col];
        const float vv = V[col];
#pragma unroll
        for (int r = 0; r < 8; ++r)
            part[r] += tanhf(acc[r] + qv) * vv;
    }
    // reduce over the 16 lanes of each half-wave (masks <16 stay in-half)
#pragma unroll
    for (int m = 8; m >= 1; m >>= 1)
#pragma unroll
        for (int r = 0; r < 8; ++r)
            part[r] += __shfl_xor(part[r], m, 32);
    if (lo == 0)
#pragma unroll
        for (int r = 0; r < 8; ++r)
            scores[row0 + hi * 8 + r] = part[r];
}

// ------------- softmax over S, context = sum_s w*enc, x = [emb(token)|context]
__global__ void ctx_kernel(const float* __restrict__ scores, const float* __restrict__ enc,
                           const int* __restrict__ token, const float* __restrict__ emb_table,
                           float* __restrict__ x) {
    __shared__ float w[S_];
    __shared__ float red[256];
    const int b = blockIdx.x, tid = threadIdx.x;
    float m = -1e30f;
    for (int s = tid; s < S_; s += 256) m = fmaxf(m, scores[b * S_ + s]);
    red[tid] = m; __syncthreads();
    for (int o = 128; o >= 1; o >>= 1) { if (tid < o) red[tid] = fmaxf(red[tid], red[tid + o]); __syncthreads(); }
    m = red[0]; __syncthreads();
    float sum = 0.0f;
    for (int s = tid; s < S_; s += 256) { float e = expf(scores[b * S_ + s] - m); w[s] = e; sum += e; }
    red[tid] = sum; __syncthreads();
    for (int o = 128; o >= 1; o >>= 1) { if (tid < o) red[tid] += red[tid + o]; __syncthreads(); }
    const float inv = 1.0f / red[0]; __syncthreads();
    for (int s = tid; s < S_; s += 256) w[s] *= inv;
    __syncthreads();
    const int tok = token[b];
    for (int e = tid; e < EMB_; e += 256)
        x[b * IN_ + e] = emb_table[(size_t)tok * EMB_ + e];
    for (int d = tid; d < DENC_; d += 256) {
        float acc = 0.0f;
        for (int s = 0; s < S_; ++s)
            acc += w[s] * enc[((size_t)b * S_ + s) * DENC_ + d];
        x[b * IN_ + EMB_ + d] = acc;
    }
}

// ------------------------------------- z = [x|h] @ [Wx;Wh] + bias  (WMMA f32)
__global__ void lstm_gemm_kernel(const float* __restrict__ x,  const float* __restrict__ h,
                                 const float* __restrict__ Wx, const float* __restrict__ Wh,
                                 const float* __restrict__ bias, float* __restrict__ z) {
    const int nt = blockIdx.x, mt = blockIdx.y;    // 256 x 4
    const int lane = threadIdx.x;
    const int lo = lane & 15, hi = lane >> 4;
    const int col = nt * 16 + lo;
    const int row = mt * 16 + lo;
    v8f acc = {};
    for (int k0 = 0; k0 < IN_; k0 += 4) {          // x part of K
        const int ka = k0 + 2 * hi;
        v2f a, bb;
        a.x  = x[row * IN_ + ka];
        a.y  = x[row * IN_ + ka + 1];
        bb.x = Wx[(size_t)ka * (4 * H_) + col];
        bb.y = Wx[(size_t)(ka + 1) * (4 * H_) + col];
        acc = __builtin_amdgcn_wmma_f32_16x16x4_f32(false, a, false, bb, (short)0, acc, false, false);
    }
    for (int k0 = 0; k0 < H_; k0 += 4) {           // h part of K
        const int ka = k0 + 2 * hi;
        v2f a, bb;
        a.x  = h[row * H_ + ka];
        a.y  = h[row * H_ + ka + 1];
        bb.x = Wh[(size_t)ka * (4 * H_) + col];
        bb.y = Wh[(size_t)(ka + 1) * (4 * H_) + col];
        acc = __builtin_amdgcn_wmma_f32_16x16x4_f32(false, a, false, bb, (short)0, acc, false, false);
    }
    const float bn = bias[col];
#pragma unroll
    for (int r = 0; r < 8; ++r) {
        const int m = mt * 16 + hi * 8 + r;
        z[(size_t)m * (4 * H_) + col] = acc[r] + bn;
    }
}

// ------------------------------------------------------------ LSTM gate math
__global__ void gates_kernel(const float* __restrict__ z, const float* __restrict__ c,
                             float* __restrict__ hn_ws, float* __restrict__ hn_out,
                             float* __restrict__ cn_out) {
    const int idx = blockIdx.x * blockDim.x + threadIdx.x;
    if (idx >= B_ * H_) return;
    const int b = idx / H_, j = idx % H_;
    const float* zr = z + (size_t)b * (4 * H_);
    const float zi = zr[j], zf = zr[H_ + j], zg = zr[2 * H_ + j], zo = zr[3 * H_ + j];
    const float cn = sigm_(zf) * c[idx] + sigm_(zi) * tanhf(zg);
    const float hn = sigm_(zo) * tanhf(cn);
    hn_ws[idx] = hn; hn_out[idx] = hn; cn_out[idx] = cn;
}

// ---------- logits = h_new @ Wfc + bfc  (WMMA f32; Wfc tile DMAed once by TDM,
//            shared by 4 waves = 4 M-tiles -> Wfc streamed from HBM exactly once)
__global__ void fc_kernel(const float* __restrict__ hn, const float* __restrict__ Wfc,
                          const float* __restrict__ bfc, float* __restrict__ logits) {
    __shared__ float lB[H_ * 16];                  // 64 KB: B tile, K-major lB[k*16 + n]
    const int nt   = blockIdx.x;                   // 2000 N-tiles
    const int tid  = threadIdx.x;                  // 128 = 4 waves
    const int wave = tid >> 5;                     // M-tile index 0..3
    const int lane = tid & 31;

    // TDM: DMA the 1024x16 f32 Wfc tile (row stride 32000) into LDS.
    if (tid == 0) {
        tdm_load_f32_tile(Wfc + (size_t)nt * 16,
                          /*tensor_d0=*/VOCAB_, /*tensor_d1=*/H_,
                          /*tile_d0=*/16,       /*tile_d1=*/H_,
                          /*stride0=*/VOCAB_);
        __builtin_amdgcn_s_wait_tensorcnt(0);
    }
    __syncthreads();

    const int lo = lane & 15, hi = lane >> 4;
    const int col = nt * 16 + lo;
    const int row = wave * 16 + lo;
    v8f acc = {};
    for (int k0 = 0; k0 < H_; k0 += 4) {
        const int ka = k0 + 2 * hi;
        v2f a, bb;
        a.x  = hn[row * H_ + ka];
        a.y  = hn[row * H_ + ka + 1];
        bb.x = lB[ka * 16 + lo];
        bb.y = lB[(ka + 1) * 16 + lo];
        acc = __builtin_amdgcn_wmma_f32_16x16x4_f32(false, a, false, bb, (short)0, acc, false, false);
    }
    const float bn = bfc[col];
#pragma unroll
    for (int r = 0; r < 8; ++r) {
        const int m = wave * 16 + hi * 8 + r;
        logits[(size_t)m * VOCAB_ + col] = acc[r] + bn;
    }
}

// ----------------------------------------------------------- vocab softmax
__global__ void vsm_kernel(const float* __restrict__ logits, float* __restrict__ probs) {
    __shared__ float red[256];
    const int b = blockIdx.x, tid = threadIdx.x;
    const float* lr = logits + (size_t)b * VOCAB_;
    float m = -1e30f;
    for (int v = tid; v < VOCAB_; v += 256) m = fmaxf(m, lr[v]);
    red[tid] = m; __syncthreads();
    for (int o = 128; o >= 1; o >>= 1) { if (tid < o) red[tid] = fmaxf(red[tid], red[tid + o]); __syncthreads(); }
    m = red[0]; __syncthreads();
    float sum = 0.0f;
    for (int v = tid; v < VOCAB_; v += 256) sum += expf(lr[v] - m);
    red[tid] = sum; __syncthreads();
    for (int o = 128; o >= 1; o >>= 1) { if (tid < o) red[tid] += red[tid + o]; __syncthreads(); }
    const float inv = 1.0f / red[0];
    float* pr = probs + (size_t)b * VOCAB_;
    for (int v = tid; v < VOCAB_; v += 256) pr[v] = expf(lr[v] - m) * inv;
}

extern "C" void kernel_launch(void* const* d_in, const int* in_sizes, int n_in,
                              void* d_out, int out_size, void* d_ws, size_t ws_size,
                              hipStream_t stream) {
    const int*   token = (const int*)d_in[0];
    const float* enc   = (const float*)d_in[1];
    const float* h     = (const float*)d_in[2];
    const float* c     = (const float*)d_in[3];
    const float* emb   = (const float*)d_in[4];
    const float* W1    = (const float*)d_in[5];
    const float* W2    = (const float*)d_in[6];
    const float* V     = (const float*)d_in[7];
    const float* Wx    = (const float*)d_in[8];
    const float* Wh    = (const float*)d_in[9];
    const float* bias  = (const float*)d_in[10];
    const float* Wfc   = (const float*)d_in[11];
    const float* bfc   = (const float*)d_in[12];

    float* out    = (float*)d_out;
    float* probs  = out;                          // [64, 32000]
    float* hn_out = out + (size_t)B_ * VOCAB_;    // [64, 1024]
    float* cn_out = hn_out + B_ * H_;             // [64, 1024]

    float* ws = (float*)d_ws;
    float* q      = ws;  ws += B_ * UNITS_;       // 32768
    float* scores = ws;  ws += B_ * S_;           // 8192
    float* x      = ws;  ws += B_ * IN_;          // 81920
    float* z      = ws;  ws += B_ * 4 * H_;       // 262144
    float* hn_ws  = ws;  ws += B_ * H_;           // 65536
    float* logits = ws;  ws += (size_t)B_ * VOCAB_; // 2048000

    q_proj_kernel  <<<B_, UNITS_, 0, stream>>>(h, W2, q);
    attn_score_kernel<<<(B_ * S_) / 16, 32, 0, stream>>>(enc, W1, V, q, scores);
    ctx_kernel     <<<B_, 256, 0, stream>>>(scores, enc, token, emb, x);
    lstm_gemm_kernel<<<dim3(4 * H_ / 16, B_ / 16), 32, 0, stream>>>(x, h, Wx, Wh, bias, z);
    gates_kernel   <<<(B_ * H_ + 255)/256, 256, 0, stream>>>(z, c, hn_ws, hn_out, cn_out);
    fc_kernel      <<<VOCAB_ / 16, 128, 0, stream>>>(hn_ws, Wfc, bfc, logits);
    vsm_kernel     <<<B_, 256, 0, stream>>>(logits, probs);
}